// HeadDiffConv_34050500722888
// MI455X (gfx1250) — compile-verified
//
#include <hip/hip_runtime.h>
#include <math.h>

#define NPTS   4096
#define KNBR   20
#define BASE_R 0.1f
#define INV2H2 50.0f               /* 1/(2*0.1*0.1) */
#define BN_S   0.9999950000374997f /* 1/sqrt(1+1e-5) */

typedef float v2f __attribute__((ext_vector_type(2)));
typedef float v8f __attribute__((ext_vector_type(8)));

// ---------------------------------------------------------------------------
// WMMA helpers: Gram-matrix tile of 16 x 16 dot products over K=4 (x,y,z,0)
// A (16x4 f32, 2 VGPR): lanes 0-15 hold {x,y} of point M=lane (K=0,1),
//                       lanes 16-31 hold {z,0} of point M=lane-16 (K=2,3).
// B (4x16 f32, 2 VGPR): same striping (rows K striped across lane halves).
// C/D (16x16 f32, 8 VGPR): element (M = v + 8*(lane>>4), N = lane&15).
// ---------------------------------------------------------------------------
__device__ __forceinline__ v2f frag_xyz(const float* __restrict__ xb, int base, int lane) {
  int p = base + (lane & 15);
  bool lo = (lane < 16);
  v2f f;
  f.x = lo ? xb[p] : xb[2 * NPTS + p];
  f.y = lo ? xb[NPTS + p] : 0.0f;
  return f;
}

__device__ __forceinline__ v8f tile_dot(v2f a, v2f b) {
  v8f c;
#pragma unroll
  for (int i = 0; i < 8; ++i) c[i] = 0.0f;
  c = __builtin_amdgcn_wmma_f32_16x16x4_f32(false, a, false, b, (short)0, c,
                                            false, false);
  return c;
}

// sum across the 16 lanes of each half-wave (lane groups 0-15 / 16-31)
__device__ __forceinline__ float red16(float x) {
  x += __shfl_xor(x, 1, 32);
  x += __shfl_xor(x, 2, 32);
  x += __shfl_xor(x, 4, 32);
  x += __shfl_xor(x, 8, 32);
  return x;
}

__device__ __forceinline__ float gelu_f(float x) {
  return 0.5f * x * (1.0f + erff(x * 0.7071067811865475f));
}

// fast exp: v_mul + v_exp_f32 (args here are <= ~10 or strongly negative)
__device__ __forceinline__ float fexp(float x) { return __expf(x); }

// ---------------------------------------------------------------------------
// P0: per-point scalars: sq = |p|^2 ; q,k projections of cat(xyz,xyz) + BN
// ---------------------------------------------------------------------------
__global__ void k_setup(const float* __restrict__ xyz,
                        const float* __restrict__ w_q, const float* __restrict__ g_q,
                        const float* __restrict__ b_q,
                        const float* __restrict__ w_k, const float* __restrict__ g_k,
                        const float* __restrict__ b_k,
                        float* __restrict__ sq, float* __restrict__ qv,
                        float* __restrict__ kv, int Bv) {
  int t = blockIdx.x * blockDim.x + threadIdx.x;
  if (t >= Bv * NPTS) return;
  int b = t / NPTS, n = t % NPTS;
  const float* xb = xyz + (size_t)b * 3 * NPTS;
  float x = xb[n], y = xb[NPTS + n], z = xb[2 * NPTS + n];
  sq[t] = x * x + y * y + z * z;
  float q = (w_q[0] + w_q[3]) * x + (w_q[1] + w_q[4]) * y + (w_q[2] + w_q[5]) * z;
  qv[t] = q * (g_q[0] * BN_S) + b_q[0];
  float k = (w_k[0] + w_k[3]) * x + (w_k[1] + w_k[4]) * y + (w_k[2] + w_k[5]) * z;
  kv[t] = k * (g_k[0] * BN_S) + b_k[0];
}

// ---------------------------------------------------------------------------
// P1: kd[n] = sum_m exp(-dist(n,m) / (2 h^2))      (row pass, WMMA)
// ---------------------------------------------------------------------------
__global__ void k_kd(const float* __restrict__ xyz, const float* __restrict__ sq,
                     float* __restrict__ kd) {
  const int b = blockIdx.y;
  const int lane = threadIdx.x & 31;
  const int wv = threadIdx.x >> 5;
  const int rowbase = blockIdx.x * 128 + wv * 16;
  const size_t bo = (size_t)b * NPTS;
  const float* xb = xyz + (size_t)b * 3 * NPTS;
  const int hi = lane >> 4, col = lane & 15;
  v2f a = frag_xyz(xb, rowbase, lane);
  float rs[8], acc[8];
#pragma unroll
  for (int v = 0; v < 8; ++v) {
    rs[v] = sq[bo + rowbase + v + 8 * hi];
    acc[v] = 0.0f;
  }
#pragma unroll 2
  for (int cb = 0; cb < NPTS; cb += 16) {
    v2f bf = frag_xyz(xb, cb, lane);
    v8f c = tile_dot(a, bf);
    float cs = sq[bo + cb + col];
#pragma unroll
    for (int v = 0; v < 8; ++v) {
      float d = rs[v] + cs - 2.0f * c[v];
      acc[v] += fexp(-d * INV2H2);
    }
  }
#pragma unroll
  for (int v = 0; v < 8; ++v) {
    float s = red16(acc[v]);
    if (col == 0) kd[bo + rowbase + v + 8 * hi] = s;
  }
}

// ---------------------------------------------------------------------------
// P2: kdmax[b] = max_n kd[b,n]
// ---------------------------------------------------------------------------
__global__ void k_max(const float* __restrict__ kd, float* __restrict__ kdmax) {
  __shared__ float sh[256];
  int b = blockIdx.x;
  float m = -3.0e38f;
  for (int i = threadIdx.x; i < NPTS; i += 256) m = fmaxf(m, kd[(size_t)b * NPTS + i]);
  sh[threadIdx.x] = m;
  __syncthreads();
  for (int o = 128; o > 0; o >>= 1) {
    if (threadIdx.x < o) sh[threadIdx.x] = fmaxf(sh[threadIdx.x], sh[threadIdx.x + o]);
    __syncthreads();
  }
  if (threadIdx.x == 0) kdmax[b] = sh[0];
}

// P3: radius = BASE_R + BASE_R * kd / (kdmax + 1e-9)
__global__ void k_radius(const float* __restrict__ kd, const float* __restrict__ kdmax,
                         float* __restrict__ rad, int Bv) {
  int t = blockIdx.x * blockDim.x + threadIdx.x;
  if (t >= Bv * NPTS) return;
  int b = t / NPTS;
  rad[t] = BASE_R + BASE_R * kd[t] / (kdmax[b] + 1e-9f);
}

// ---------------------------------------------------------------------------
// P4: sinv[n] = 1 / sum_{m : dist<rad_n} exp(q_n k_m)   (row pass, WMMA)
// ---------------------------------------------------------------------------
__global__ void k_srow(const float* __restrict__ xyz, const float* __restrict__ sq,
                       const float* __restrict__ qv, const float* __restrict__ kv,
                       const float* __restrict__ rad, float* __restrict__ sinv) {
  const int b = blockIdx.y;
  const int lane = threadIdx.x & 31;
  const int wv = threadIdx.x >> 5;
  const int rowbase = blockIdx.x * 128 + wv * 16;
  const size_t bo = (size_t)b * NPTS;
  const float* xb = xyz + (size_t)b * 3 * NPTS;
  const int hi = lane >> 4, col = lane & 15;
  v2f a = frag_xyz(xb, rowbase, lane);
  float rs[8], qr[8], rr[8], acc[8];
#pragma unroll
  for (int v = 0; v < 8; ++v) {
    int r = rowbase + v + 8 * hi;
    rs[v] = sq[bo + r]; qr[v] = qv[bo + r]; rr[v] = rad[bo + r]; acc[v] = 0.0f;
  }
#pragma unroll 2
  for (int cb = 0; cb < NPTS; cb += 16) {
    v2f bf = frag_xyz(xb, cb, lane);
    v8f c = tile_dot(a, bf);
    float cs = sq[bo + cb + col];
    float kc = kv[bo + cb + col];
#pragma unroll
    for (int v = 0; v < 8; ++v) {
      float d = rs[v] + cs - 2.0f * c[v];
      acc[v] += (d < rr[v]) ? fexp(qr[v] * kc) : 0.0f;
    }
  }
#pragma unroll
  for (int v = 0; v < 8; ++v) {
    float s = red16(acc[v]);
    if (col == 0) sinv[bo + rowbase + v + 8 * hi] = 1.0f / s;
  }
}

// ---------------------------------------------------------------------------
// P5: cinv[m] = 1 / sum_n adj0[n,m]                     (column pass, WMMA)
// ---------------------------------------------------------------------------
__global__ void k_csum(const float* __restrict__ xyz, const float* __restrict__ sq,
                       const float* __restrict__ qv, const float* __restrict__ kv,
                       const float* __restrict__ rad, const float* __restrict__ sinv,
                       float* __restrict__ cinv) {
  const int b = blockIdx.y;
  const int lane = threadIdx.x & 31;
  const int wv = threadIdx.x >> 5;
  const int mbase = blockIdx.x * 128 + wv * 16;
  const size_t bo = (size_t)b * NPTS;
  const float* xb = xyz + (size_t)b * 3 * NPTS;
  const int hi = lane >> 4, col = lane & 15;
  v2f a = frag_xyz(xb, mbase, lane);
  float sm[8], km[8], acc[8];
#pragma unroll
  for (int v = 0; v < 8; ++v) {
    int m = mbase + v + 8 * hi;
    sm[v] = sq[bo + m]; km[v] = kv[bo + m]; acc[v] = 0.0f;
  }
#pragma unroll 2
  for (int nb = 0; nb < NPTS; nb += 16) {
    v2f bf = frag_xyz(xb, nb, lane);
    v8f c = tile_dot(a, bf);
    int n = nb + col;
    float sn = sq[bo + n], qn = qv[bo + n], rn = rad[bo + n], si = sinv[bo + n];
#pragma unroll
    for (int v = 0; v < 8; ++v) {
      float d = sm[v] + sn - 2.0f * c[v];
      acc[v] += (d < rn) ? fexp(qn * km[v]) * si : 0.0f;
    }
  }
#pragma unroll
  for (int v = 0; v < 8; ++v) {
    float s = red16(acc[v]);
    if (col == 0) cinv[bo + mbase + v + 8 * hi] = 1.0f / s;
  }
}

// ---------------------------------------------------------------------------
// P6: rinv[n] = 1 / ( sinv[n] * sum_m keep*exp(q_n k_m)*cinv[m] )  (row pass)
// ---------------------------------------------------------------------------
__global__ void k_rinv(const float* __restrict__ xyz, const float* __restrict__ sq,
                       const float* __restrict__ qv, const float* __restrict__ kv,
                       const float* __restrict__ rad, const float* __restrict__ cinv,
                       const float* __restrict__ sinv, float* __restrict__ rinv) {
  const int b = blockIdx.y;
  const int lane = threadIdx.x & 31;
  const int wv = threadIdx.x >> 5;
  const int rowbase = blockIdx.x * 128 + wv * 16;
  const size_t bo = (size_t)b * NPTS;
  const float* xb = xyz + (size_t)b * 3 * NPTS;
  const int hi = lane >> 4, col = lane & 15;
  v2f a = frag_xyz(xb, rowbase, lane);
  float rs[8], qr[8], rr[8], acc[8];
#pragma unroll
  for (int v = 0; v < 8; ++v) {
    int r = rowbase + v + 8 * hi;
    rs[v] = sq[bo + r]; qr[v] = qv[bo + r]; rr[v] = rad[bo + r]; acc[v] = 0.0f;
  }
#pragma unroll 2
  for (int cb = 0; cb < NPTS; cb += 16) {
    v2f bf = frag_xyz(xb, cb, lane);
    v8f c = tile_dot(a, bf);
    float cs = sq[bo + cb + col];
    float kc = kv[bo + cb + col];
    float ci = cinv[bo + cb + col];
#pragma unroll
    for (int v = 0; v < 8; ++v) {
      float d = rs[v] + cs - 2.0f * c[v];
      acc[v] += (d < rr[v]) ? fexp(qr[v] * kc) * ci : 0.0f;
    }
  }
#pragma unroll
  for (int v = 0; v < 8; ++v) {
    float s = red16(acc[v]);
    if (col == 0) {
      int r = rowbase + v + 8 * hi;
      rinv[bo + r] = 1.0f / (s * sinv[bo + r]);
    }
  }
}

// P6b: wx[b,ch,n] = xyz * sinv * rinv
__global__ void k_wx(const float* __restrict__ xyz, const float* __restrict__ sinv,
                     const float* __restrict__ rinv, float* __restrict__ wx, int Bv) {
  int t = blockIdx.x * blockDim.x + threadIdx.x;
  if (t >= Bv * NPTS) return;
  int b = t / NPTS, n = t % NPTS;
  float f = sinv[t] * rinv[t];
  const float* xb = xyz + (size_t)b * 3 * NPTS;
  float* wb = wx + (size_t)b * 3 * NPTS;
  wb[n] = xb[n] * f;
  wb[NPTS + n] = xb[NPTS + n] * f;
  wb[2 * NPTS + n] = xb[2 * NPTS + n] * f;
}

// ---------------------------------------------------------------------------
// P7: smoothed[ch,m] = cinv[m] * sum_n wx[ch,n]*keep*exp(q_n k_m)  (col pass)
// ---------------------------------------------------------------------------
__global__ void k_smooth(const float* __restrict__ xyz, const float* __restrict__ sq,
                         const float* __restrict__ qv, const float* __restrict__ kv,
                         const float* __restrict__ rad, const float* __restrict__ wx,
                         const float* __restrict__ cinv, float* __restrict__ smooth) {
  const int b = blockIdx.y;
  const int lane = threadIdx.x & 31;
  const int wv = threadIdx.x >> 5;
  const int mbase = blockIdx.x * 128 + wv * 16;
  const size_t bo = (size_t)b * NPTS;
  const float* xb = xyz + (size_t)b * 3 * NPTS;
  const float* wb = wx + (size_t)b * 3 * NPTS;
  const int hi = lane >> 4, col = lane & 15;
  v2f a = frag_xyz(xb, mbase, lane);
  float sm[8], km[8], a0[8], a1[8], a2[8];
#pragma unroll
  for (int v = 0; v < 8; ++v) {
    int m = mbase + v + 8 * hi;
    sm[v] = sq[bo + m]; km[v] = kv[bo + m];
    a0[v] = 0.0f; a1[v] = 0.0f; a2[v] = 0.0f;
  }
#pragma unroll 2
  for (int nb = 0; nb < NPTS; nb += 16) {
    v2f bf = frag_xyz(xb, nb, lane);
    v8f c = tile_dot(a, bf);
    int n = nb + col;
    float sn = sq[bo + n], qn = qv[bo + n], rn = rad[bo + n];
    float w0 = wb[n], w1 = wb[NPTS + n], w2 = wb[2 * NPTS + n];
#pragma unroll
    for (int v = 0; v < 8; ++v) {
      float d = sm[v] + sn - 2.0f * c[v];
      if (d < rn) {
        float e = fexp(qn * km[v]);
        a0[v] += e * w0; a1[v] += e * w1; a2[v] += e * w2;
      }
    }
  }
#pragma unroll
  for (int v = 0; v < 8; ++v) {
    float s0 = red16(a0[v]);
    float s1 = red16(a1[v]);
    float s2 = red16(a2[v]);
    if (col == 0) {
      int m = mbase + v + 8 * hi;
      float ci = cinv[bo + m];
      float* ob = smooth + (size_t)b * 3 * NPTS;
      ob[m] = s0 * ci;
      ob[NPTS + m] = s1 * ci;
      ob[2 * NPTS + m] = s2 * ci;
    }
  }
}

// ---------------------------------------------------------------------------
// P8: per-row ball query: first KNBR ascending indices with dist <= radius
// (wave per row; ballot scan with early exit; pad with first neighbor)
// ---------------------------------------------------------------------------
__global__ void k_nbr(const float* __restrict__ xyz, const float* __restrict__ rad,
                      int* __restrict__ nbr, int Bv) {
  int gw = (blockIdx.x * blockDim.x + threadIdx.x) >> 5;
  int lane = threadIdx.x & 31;
  int b = gw / NPTS, n = gw % NPTS;
  if (b >= Bv) return;
  const float* xb = xyz + (size_t)b * 3 * NPTS;
  float px = xb[n], py = xb[NPTS + n], pz = xb[2 * NPTS + n];
  float r = rad[(size_t)b * NPTS + n];
  int cnt = 0;
  int loc[KNBR];
  for (int base = 0; base < NPTS && cnt < KNBR; base += 32) {
    int m = base + lane;
    float dx = xb[m] - px, dy = xb[NPTS + m] - py, dz = xb[2 * NPTS + m] - pz;
    float d = dx * dx + dy * dy + dz * dz;
    unsigned msk = (unsigned)__ballot(d <= r);
    if (lane == 0) {
      while (msk && cnt < KNBR) {
        int t = __builtin_ctz(msk);
        msk &= msk - 1;
        loc[cnt++] = base + t;
      }
    }
    cnt = __shfl(cnt, 0, 32);
  }
  if (lane == 0) {
    if (cnt == 0) { loc[0] = n; cnt = 1; }
    for (int i = cnt; i < KNBR; ++i) loc[i] = loc[0];
    int* o = nbr + (size_t)((size_t)b * NPTS + n) * KNBR;
    for (int i = 0; i < KNBR; ++i) o[i] = loc[i];
  }
}

// ---------------------------------------------------------------------------
// P9: fused per-point tail: PE gather + conv9->8 + BN + GELU + max_k,
// 8->16 (BN,GELU), 16->64 (BN) = pos ; v = BN(w_v * [variation, xyz]) ;
// out = BN(w_out * gelu(v + pos))
// ---------------------------------------------------------------------------
__global__ void k_final(const float* __restrict__ xyz, const float* __restrict__ smooth,
                        const int* __restrict__ nbr,
                        const float* __restrict__ w_v, const float* __restrict__ g_v,
                        const float* __restrict__ b_v,
                        const float* __restrict__ w_pe, const float* __restrict__ b_pe,
                        const float* __restrict__ g_pe, const float* __restrict__ bb_pe,
                        const float* __restrict__ w_m1, const float* __restrict__ g_m1,
                        const float* __restrict__ b_m1,
                        const float* __restrict__ w_m2, const float* __restrict__ g_m2,
                        const float* __restrict__ b_m2,
                        const float* __restrict__ w_out, const float* __restrict__ g_out,
                        const float* __restrict__ b_out,
                        float* __restrict__ out, int Bv) {
  __shared__ float s_wpe[72], s_bpe[8], s_gpe[8], s_b2[8];
  __shared__ float s_wm1[128], s_gm1[16], s_bm1[16];
  __shared__ float s_wm2[1024], s_gm2[64], s_bm2[64];
  __shared__ float s_wv[384], s_gv[64], s_bv[64];
  __shared__ float s_wo[2048], s_go[32], s_bo[32];
  const int tid = threadIdx.x, nt = blockDim.x;
  for (int i = tid; i < 72; i += nt) s_wpe[i] = w_pe[i];
  for (int i = tid; i < 8; i += nt) { s_bpe[i] = b_pe[i]; s_gpe[i] = g_pe[i]; s_b2[i] = bb_pe[i]; }
  for (int i = tid; i < 128; i += nt) s_wm1[i] = w_m1[i];
  for (int i = tid; i < 16; i += nt) { s_gm1[i] = g_m1[i]; s_bm1[i] = b_m1[i]; }
  for (int i = tid; i < 1024; i += nt) s_wm2[i] = w_m2[i];
  for (int i = tid; i < 64; i += nt) {
    s_gm2[i] = g_m2[i]; s_bm2[i] = b_m2[i]; s_gv[i] = g_v[i]; s_bv[i] = b_v[i];
  }
  for (int i = tid; i < 384; i += nt) s_wv[i] = w_v[i];
  for (int i = tid; i < 2048; i += nt) s_wo[i] = w_out[i];
  for (int i = tid; i < 32; i += nt) { s_go[i] = g_out[i]; s_bo[i] = b_out[i]; }
  __syncthreads();

  int t = blockIdx.x * blockDim.x + tid;
  if (t >= Bv * NPTS) return;
  int b = t / NPTS, n = t % NPTS;
  const float* xb = xyz + (size_t)b * 3 * NPTS;
  float px = xb[n], py = xb[NPTS + n], pz = xb[2 * NPTS + n];

  // ---- position encoder: gather K neighbors, conv(9->8)+BN+GELU, max ----
  const int* nb = nbr + (size_t)t * KNBR;
  float pf8[8];
#pragma unroll
  for (int oc = 0; oc < 8; ++oc) pf8[oc] = -3.0e38f;
  for (int k = 0; k < KNBR; ++k) {
    int m = nb[k];
    float gx = xb[m], gy = xb[NPTS + m], gz = xb[2 * NPTS + m];
    float f9[9] = {px, py, pz, gx, gy, gz, gx - px, gy - py, gz - pz};
#pragma unroll
    for (int oc = 0; oc < 8; ++oc) {
      float acc = s_bpe[oc];
#pragma unroll
      for (int c = 0; c < 9; ++c) acc += s_wpe[oc * 9 + c] * f9[c];
      acc = acc * (s_gpe[oc] * BN_S) + s_b2[oc];
      acc = gelu_f(acc);
      pf8[oc] = fmaxf(pf8[oc], acc);
    }
  }
  float pf16[16];
#pragma unroll
  for (int oc = 0; oc < 16; ++oc) {
    float acc = 0.0f;
#pragma unroll
    for (int c = 0; c < 8; ++c) acc += s_wm1[oc * 8 + c] * pf8[c];
    pf16[oc] = gelu_f(acc * (s_gm1[oc] * BN_S) + s_bm1[oc]);
  }

  // ---- value branch: variation = smoothed - xyz ----
  const float* sb = smooth + (size_t)b * 3 * NPTS;
  float f6[6] = {sb[n] - px, sb[NPTS + n] - py, sb[2 * NPTS + n] - pz, px, py, pz};

  float o64[64];
#pragma unroll
  for (int oc = 0; oc < 64; ++oc) {
    float pos = 0.0f;
#pragma unroll
    for (int c = 0; c < 16; ++c) pos += s_wm2[oc * 16 + c] * pf16[c];
    pos = pos * (s_gm2[oc] * BN_S) + s_bm2[oc];
    float vv = 0.0f;
#pragma unroll
    for (int c = 0; c < 6; ++c) vv += s_wv[oc * 6 + c] * f6[c];
    vv = vv * (s_gv[oc] * BN_S) + s_bv[oc];
    o64[oc] = gelu_f(vv + pos);
  }
#pragma unroll
  for (int oc = 0; oc < 32; ++oc) {
    float acc = 0.0f;
#pragma unroll
    for (int c = 0; c < 64; ++c) acc += s_wo[oc * 64 + c] * o64[c];
    acc = acc * (s_go[oc] * BN_S) + s_bo[oc];
    out[((size_t)b * 32 + oc) * NPTS + n] = acc;
  }
}

// ---------------------------------------------------------------------------
extern "C" void kernel_launch(void* const* d_in, const int* in_sizes, int n_in,
                              void* d_out, int out_size, void* d_ws, size_t ws_size,
                              hipStream_t stream) {
  const float* xyz  = (const float*)d_in[0];
  const float* w_q  = (const float*)d_in[1];
  const float* g_q  = (const float*)d_in[2];
  const float* b_q  = (const float*)d_in[3];
  const float* w_k  = (const float*)d_in[4];
  const float* g_k  = (const float*)d_in[5];
  const float* b_k  = (const float*)d_in[6];
  const float* w_v  = (const float*)d_in[7];
  const float* g_v  = (const float*)d_in[8];
  const float* b_v  = (const float*)d_in[9];
  const float* w_pe = (const float*)d_in[10];
  const float* b_pe = (const float*)d_in[11];
  const float* g_pe = (const float*)d_in[12];
  const float* bb_pe= (const float*)d_in[13];
  const float* w_m1 = (const float*)d_in[14];
  const float* g_m1 = (const float*)d_in[15];
  const float* b_m1 = (const float*)d_in[16];
  const float* w_m2 = (const float*)d_in[17];
  const float* g_m2 = (const float*)d_in[18];
  const float* b_m2 = (const float*)d_in[19];
  const float* w_out= (const float*)d_in[20];
  const float* g_out= (const float*)d_in[21];
  const float* b_out= (const float*)d_in[22];

  const int Bv = in_sizes[0] / (3 * NPTS);
  const int BN = Bv * NPTS;

  // workspace layout
  float* ws = (float*)d_ws;
  float* sq    = ws;  ws += BN;
  float* qv    = ws;  ws += BN;
  float* kv    = ws;  ws += BN;
  float* kd    = ws;  ws += BN;
  float* rad   = ws;  ws += BN;
  float* sinv  = ws;  ws += BN;
  float* cinv  = ws;  ws += BN;
  float* rinv  = ws;  ws += BN;
  float* wx    = ws;  ws += 3 * BN;
  float* smo   = ws;  ws += 3 * BN;
  float* kdmax = ws;  ws += Bv + 4;
  int*   nbr   = (int*)ws;

  float* out = (float*)d_out;

  dim3 blk256(256);
  dim3 gPoint((BN + 255) / 256);
  dim3 gTile(NPTS / 128, Bv);

  k_setup<<<gPoint, blk256, 0, stream>>>(xyz, w_q, g_q, b_q, w_k, g_k, b_k,
                                         sq, qv, kv, Bv);
  k_kd<<<gTile, blk256, 0, stream>>>(xyz, sq, kd);
  k_max<<<dim3(Bv), blk256, 0, stream>>>(kd, kdmax);
  k_radius<<<gPoint, blk256, 0, stream>>>(kd, kdmax, rad, Bv);
  k_srow<<<gTile, blk256, 0, stream>>>(xyz, sq, qv, kv, rad, sinv);
  k_csum<<<gTile, blk256, 0, stream>>>(xyz, sq, qv, kv, rad, sinv, cinv);
  k_rinv<<<gTile, blk256, 0, stream>>>(xyz, sq, qv, kv, rad, cinv, sinv, rinv);
  k_wx<<<gPoint, blk256, 0, stream>>>(xyz, sinv, rinv, wx, Bv);
  k_smooth<<<gTile, blk256, 0, stream>>>(xyz, sq, qv, kv, rad, wx, cinv, smo);
  k_nbr<<<dim3(BN / 8), blk256, 0, stream>>>(xyz, rad, nbr, Bv);
  k_final<<<dim3(BN / 128), dim3(128), 0, stream>>>(
      xyz, smo, nbr, w_v, g_v, b_v, w_pe, b_pe, g_pe, bb_pe,
      w_m1, g_m1, b_m1, w_m2, g_m2, b_m2, w_out, g_out, b_out, out, Bv);
}